// EpisodicMemory_55027120996865
// MI455X (gfx1250) — compile-verified
//
#include <hip/hip_runtime.h>
#include <hip/hip_bf16.h>
#include <math.h>

typedef __attribute__((ext_vector_type(2))) float v2f;
typedef __attribute__((ext_vector_type(8))) float v8f;

#define DIMD   128
#define BQ     64
#define TOPK   16
#define TILE_N 1024
#define NT16   (TILE_N / 16)   // 64 n-tiles per block
#define NWAVES 8
#define EPSN   1e-8f
#define RW     0.3f

// ---------------------------------------------------------------------------
// Kernel 1: qhat = normalize(query @ Wq^T + bq)
// grid = 64 blocks (one per query row), 128 threads (one per output dim)
// ---------------------------------------------------------------------------
__global__ __launch_bounds__(DIMD)
void qproj_kernel(const float* __restrict__ query,
                  const float* __restrict__ Wq,
                  const float* __restrict__ bq,
                  float* __restrict__ qhat)
{
    const int b = blockIdx.x;
    const int d = threadIdx.x;
    const float* qr = query + b * DIMD;
    const float* wr = Wq + d * DIMD;   // Wq.T column d == Wq[d][*]
    float acc = bq[d];
    #pragma unroll 8
    for (int j = 0; j < DIMD; ++j)
        acc = fmaf(qr[j], wr[j], acc);

    __shared__ float sq[DIMD];
    sq[d] = acc * acc;
    __syncthreads();
    for (int off = DIMD / 2; off > 0; off >>= 1) {
        if (d < off) sq[d] += sq[d + off];
        __syncthreads();
    }
    const float qn = fmaxf(sqrtf(sq[0]), EPSN);
    qhat[b * DIMD + d] = acc / qn;
}

// ---------------------------------------------------------------------------
// Kernel 2: fused  sim = qhat @ (mem/||mem||)^T  ->  score  ->  per-block top-16
// grid = ceil(N/1024) blocks, 256 threads (8 waves of 32)
// WMMA f32 16x16x4: A/B operand layout per lane:
//   lane l holds row (l&15), k = kb + 2*(l>>4) + {0,1}   (v2f)
// C layout: vgpr r, lane l -> row = r + 8*(l>>4), col = l&15
// ---------------------------------------------------------------------------
__global__ __launch_bounds__(256)
void score_topk_kernel(const float* __restrict__ mem,
                       const int*   __restrict__ ts,
                       const float* __restrict__ imp,
                       const float* __restrict__ qhat,
                       float* __restrict__ cand_val,
                       int*   __restrict__ cand_idx,
                       const int N, const float invNp1)
{
    __shared__ float smem[BQ * TILE_N];   // 256 KB score tile (CDNA5: 320KB/WG ok)

    const int lane   = threadIdx.x & 31;
    const int wave   = threadIdx.x >> 5;
    const int col    = lane & 15;   // matrix row (B) / query row (A) within tile
    const int khalf  = lane >> 4;   // which K pair
    const int base_n = blockIdx.x * TILE_N;

    for (int nt = wave; nt < NT16; nt += NWAVES) {
        const int  n     = base_n + nt * 16 + col;
        const bool valid = (n < N);
        const int  nclmp = valid ? n : 0;
        const float* mrow = mem + (size_t)nclmp * DIMD;

        v8f c[4];
        c[0] = (v8f){}; c[1] = (v8f){}; c[2] = (v8f){}; c[3] = (v8f){};
        float sumsq = 0.f;

        #pragma unroll 4
        for (int kb = 0; kb < DIMD; kb += 4) {
            const int kc = kb + 2 * khalf;
            v2f bv = *(const v2f*)(mrow + kc);     // in-bounds (clamped row)
            if (!valid) { bv.x = 0.f; bv.y = 0.f; }
            sumsq = fmaf(bv.x, bv.x, sumsq);
            sumsq = fmaf(bv.y, bv.y, sumsq);
            #pragma unroll
            for (int mt = 0; mt < 4; ++mt) {
                v2f av = *(const v2f*)(qhat + (mt * 16 + col) * DIMD + kc);
                c[mt] = __builtin_amdgcn_wmma_f32_16x16x4_f32(
                            false, av, false, bv, (short)0, c[mt], false, false);
            }
        }

        // full row ||mem[n]||^2 : combine the two K-halves (lanes l and l^16)
        sumsq += __shfl_xor(sumsq, 16, 32);
        const float inv_mn = 1.f / fmaxf(sqrtf(sumsq), EPSN);

        const float rec  = ((float)ts[nclmp] + 1.f) * invNp1;
        const float impf = 0.5f + 0.5f * imp[nclmp];

        const int nl = nt * 16 + col;
        #pragma unroll
        for (int mt = 0; mt < 4; ++mt) {
            #pragma unroll
            for (int r = 0; r < 8; ++r) {
                float s = ((1.0f - RW) * c[mt][r] * inv_mn + RW * rec) * impf;
                if (!valid) s = -INFINITY;
                const int q = mt * 16 + khalf * 8 + r;
                smem[q * TILE_N + nl] = s;
            }
        }
    }
    __syncthreads();

    // --- per-block top-16 for each query row; wave w owns queries 8w..8w+7 ---
    float* base_cv = cand_val + (size_t)blockIdx.x * BQ * TOPK;
    int*   base_ci = cand_idx + (size_t)blockIdx.x * BQ * TOPK;

    for (int q = wave * 8; q < wave * 8 + 8; ++q) {
        float* row = smem + q * TILE_N;
        for (int it = 0; it < TOPK; ++it) {
            float bvv = -INFINITY;
            int   bj  = 0x3fffffff;
            for (int j = lane; j < TILE_N; j += 32) {
                const float v = row[j];
                if (v > bvv || (v == bvv && j < bj)) { bvv = v; bj = j; }
            }
            #pragma unroll
            for (int off = 16; off > 0; off >>= 1) {
                const float ov = __shfl_xor(bvv, off, 32);
                const int   oj = __shfl_xor(bj, off, 32);
                if (ov > bvv || (ov == bvv && oj < bj)) { bvv = ov; bj = oj; }
            }
            if (lane == 0) {
                base_cv[q * TOPK + it] = bvv;
                base_ci[q * TOPK + it] = base_n + bj;
            }
            if (bj < TILE_N) row[bj] = -INFINITY;  // all lanes, same addr/value
        }
    }
}

// ---------------------------------------------------------------------------
// Kernel 3: merge per-block candidates -> final top-16 (desc, ties -> low idx)
// grid = 64 blocks (one per query), 256 threads
// ---------------------------------------------------------------------------
__global__ __launch_bounds__(256)
void merge_topk_kernel(const int nblk,
                       float* __restrict__ cand_val,
                       const int* __restrict__ cand_idx,
                       float* __restrict__ out_val,
                       int*   __restrict__ out_idx)
{
    const int q   = blockIdx.x;
    const int tid = threadIdx.x;
    const int M   = nblk * TOPK;

    __shared__ float sv[256];
    __shared__ int   sp[256];
    __shared__ int   sg[256];

    for (int it = 0; it < TOPK; ++it) {
        float bvv = -INFINITY;
        int   bp  = -1;
        int   bg  = 0x7fffffff;
        for (int j = tid; j < M; j += 256) {
            const int blk = j >> 4, e = j & 15;
            const int pos = (blk * BQ + q) * TOPK + e;
            const float v = cand_val[pos];
            const int   g = cand_idx[pos];
            if (v > bvv || (v == bvv && g < bg)) { bvv = v; bp = pos; bg = g; }
        }
        sv[tid] = bvv; sp[tid] = bp; sg[tid] = bg;
        __syncthreads();
        for (int off = 128; off > 0; off >>= 1) {
            if (tid < off) {
                const float ov = sv[tid + off];
                const int   og = sg[tid + off];
                if (ov > sv[tid] || (ov == sv[tid] && og < sg[tid])) {
                    sv[tid] = ov; sg[tid] = og; sp[tid] = sp[tid + off];
                }
            }
            __syncthreads();
        }
        if (tid == 0) {
            out_val[q * TOPK + it] = sv[0];
            out_idx[q * TOPK + it] = sg[0];
            if (sp[0] >= 0) cand_val[sp[0]] = -INFINITY;  // remove winner
        }
        __syncthreads();
    }
}

// ---------------------------------------------------------------------------
// inputs: 0 query[64*128] f32, 1 mem_content[1e6*128] f32, 2 timestamps[1e6] int,
//         3 importance[1e6] f32, 4 Wq[128*128] f32, 5 bq[128] f32, 6 k (int, =16)
// output: top_vals[64*16] f32 ++ top_idx[64*16] int32
// ---------------------------------------------------------------------------
extern "C" void kernel_launch(void* const* d_in, const int* in_sizes, int n_in,
                              void* d_out, int out_size, void* d_ws, size_t ws_size,
                              hipStream_t stream)
{
    const float* query = (const float*)d_in[0];
    const float* mem   = (const float*)d_in[1];
    const int*   ts    = (const int*)  d_in[2];
    const float* imp   = (const float*)d_in[3];
    const float* Wq    = (const float*)d_in[4];
    const float* bq    = (const float*)d_in[5];
    // d_in[6] (k) is on device; k == TOPK == 16 per reference setup.

    const int N    = in_sizes[2];
    const int nblk = (N + TILE_N - 1) / TILE_N;

    // workspace layout
    float* qhat     = (float*)d_ws;                                   // 32 KB
    char*  p        = (char*)d_ws + (size_t)BQ * DIMD * sizeof(float);
    float* cand_val = (float*)p;                                      // nblk*64*16 f32
    int*   cand_idx = (int*)(p + (size_t)nblk * BQ * TOPK * sizeof(float));

    float* out_val = (float*)d_out;
    int*   out_idx = (int*)d_out + BQ * TOPK;

    qproj_kernel<<<BQ, DIMD, 0, stream>>>(query, Wq, bq, qhat);
    score_topk_kernel<<<nblk, 256, 0, stream>>>(mem, ts, imp, qhat,
                                                cand_val, cand_idx,
                                                N, 1.0f / (float)(N + 1));
    merge_topk_kernel<<<BQ, 256, 0, stream>>>(nblk, cand_val, cand_idx,
                                              out_val, out_idx);
}